// MobilityPyGEncoder_38929583571345
// MI455X (gfx1250) — compile-verified
//
#include <hip/hip_runtime.h>
#include <hip/hip_bf16.h>

typedef __attribute__((ext_vector_type(2))) float v2f;
typedef __attribute__((ext_vector_type(8))) float v8f;

// ---------------------------------------------------------------------------
// Utility: fill a float buffer with a constant (grid-stride)
// ---------------------------------------------------------------------------
__global__ void fill_kernel(float* __restrict__ p, float val, size_t n) {
  size_t i = (size_t)blockIdx.x * blockDim.x + threadIdx.x;
  size_t stride = (size_t)gridDim.x * blockDim.x;
  for (; i < n; i += stride) p[i] = val;
}

// ---------------------------------------------------------------------------
// Degree accumulation: deg[dst[e]] += w[e]   (deg pre-filled with 1.0 = self loop)
// ---------------------------------------------------------------------------
__global__ void deg_kernel(const long long* __restrict__ dst,
                           const float* __restrict__ w,
                           float* __restrict__ deg, int E) {
  int e = blockIdx.x * blockDim.x + threadIdx.x;
  if (e < E) atomicAdd(&deg[(int)dst[e]], w[e]);
}

__global__ void dinv_kernel(const float* __restrict__ deg,
                            float* __restrict__ dinv, int Nn) {
  int n = blockIdx.x * blockDim.x + threadIdx.x;
  if (n < Nn) {
    float d = deg[n];
    dinv[n] = (d > 0.f) ? rsqrtf(fmaxf(d, 1e-12f)) : 0.f;
  }
}

__global__ void norm_kernel(const long long* __restrict__ src,
                            const long long* __restrict__ dst,
                            const float* __restrict__ w,
                            const float* __restrict__ dinv,
                            float* __restrict__ normv, int E) {
  int e = blockIdx.x * blockDim.x + threadIdx.x;
  if (e < E) normv[e] = dinv[(int)src[e]] * w[e] * dinv[(int)dst[e]];
}

// ---------------------------------------------------------------------------
// FP32 WMMA GEMM: C[Nrows, NC] = A[Nrows, KDIM] * B[KDIM, NC] (+ bias)
// One wave computes a 16(M) x NC strip using V_WMMA_F32_16X16X4_F32.
// 32-bit A 16x4 layout: lanes 0-15 -> {K0,K1}, lanes 16-31 -> {K2,K3}
// 32-bit C/D layout:    VGPR r -> row r (lanes 0-15) / row r+8 (lanes 16-31)
// Nrows must be a multiple of 16 (N = 50000 = 3125 * 16).
// ---------------------------------------------------------------------------
template <int KDIM, int NC>
__global__ __launch_bounds__(128) void gemm_wmma_kernel(
    const float* __restrict__ A, const float* __restrict__ B,
    const float* __restrict__ bias, float* __restrict__ C, int Nrows) {
  constexpr int NT = NC / 16;
  int tile = blockIdx.x * (blockDim.x >> 5) + (threadIdx.x >> 5);
  int lane = threadIdx.x & 31;
  int m = lane & 15;
  int hi = lane >> 4;
  int row0 = tile * 16;
  if (row0 >= Nrows) return;  // wave-uniform: EXEC stays all-ones for live waves

  v8f acc[NT];
  const v8f vzero = {0.f, 0.f, 0.f, 0.f, 0.f, 0.f, 0.f, 0.f};
#pragma unroll
  for (int j = 0; j < NT; ++j) acc[j] = vzero;

  const float* arow = A + (size_t)(row0 + m) * KDIM + 2 * hi;
#pragma unroll
  for (int k = 0; k < KDIM; k += 4) {
    v2f a = *(const v2f*)(arow + k);  // A[m][k+2*hi], A[m][k+2*hi+1]
#pragma unroll
    for (int j = 0; j < NT; ++j) {
      const float* bp = B + (size_t)(k + 2 * hi) * NC + j * 16 + m;
      v2f b;
      b.x = bp[0];    // B[k+2*hi  ][col]
      b.y = bp[NC];   // B[k+2*hi+1][col]
      acc[j] = __builtin_amdgcn_wmma_f32_16x16x4_f32(
          false, a, false, b, (short)0, acc[j], false, false);
    }
  }

#pragma unroll
  for (int j = 0; j < NT; ++j) {
    int col = j * 16 + m;
    float bv = bias ? bias[col] : 0.f;
#pragma unroll
    for (int r = 0; r < 8; ++r) {
      int row = row0 + r + 8 * hi;
      C[(size_t)row * NC + col] = acc[j][r] + bv;
    }
  }
}

// ---------------------------------------------------------------------------
// Edge scatter: acc[dst[e], f] += feat[src[e], f] * norm[e]
// One wave per edge, 3 features per lane (96 = 3*32).
// ---------------------------------------------------------------------------
__global__ __launch_bounds__(256) void scatter_kernel(
    const float* __restrict__ feat, const long long* __restrict__ src,
    const long long* __restrict__ dst, const float* __restrict__ normv,
    float* __restrict__ acc, int E) {
  int gid = blockIdx.x * blockDim.x + threadIdx.x;
  int e = gid >> 5;
  int lane = threadIdx.x & 31;
  if (e >= E) return;
  int s = (int)src[e];
  int d = (int)dst[e];
  float w = normv[e];
  const float* sp = feat + (size_t)s * 96;
  float* dp = acc + (size_t)d * 96;
#pragma unroll
  for (int j = 0; j < 3; ++j) {
    int f = lane + 32 * j;
    atomicAdd(&dp[f], sp[f] * w);
  }
}

// ---------------------------------------------------------------------------
// Layer-0 epilogue: g = acc + xW0*dinv^2 + b0 ; h = relu(LN(g)) + skip
// One wave per node; wave32 shuffle reductions. hbuf holds skip on entry,
// h on exit (in place).
// ---------------------------------------------------------------------------
__global__ __launch_bounds__(256) void post0_kernel(
    const float* __restrict__ acc, const float* __restrict__ xw,
    const float* __restrict__ dinv, const float* __restrict__ b0,
    const float* __restrict__ ln_g, const float* __restrict__ ln_b,
    float* __restrict__ hbuf, int Nn) {
  int wid = (blockIdx.x * blockDim.x + threadIdx.x) >> 5;
  int lane = threadIdx.x & 31;
  if (wid >= Nn) return;
  float di = dinv[wid];
  float di2 = di * di;
  size_t base = (size_t)wid * 96;
  float v[3];
  float sum = 0.f;
#pragma unroll
  for (int j = 0; j < 3; ++j) {
    int f = lane + 32 * j;
    float g = acc[base + f] + xw[base + f] * di2 + b0[f];
    v[j] = g;
    sum += g;
  }
#pragma unroll
  for (int off = 16; off > 0; off >>= 1) sum += __shfl_xor(sum, off, 32);
  float mu = sum * (1.0f / 96.0f);
  float var = 0.f;
#pragma unroll
  for (int j = 0; j < 3; ++j) {
    float t = v[j] - mu;
    var += t * t;
  }
#pragma unroll
  for (int off = 16; off > 0; off >>= 1) var += __shfl_xor(var, off, 32);
  var *= (1.0f / 96.0f);
  float rstd = rsqrtf(var + 1e-5f);
#pragma unroll
  for (int j = 0; j < 3; ++j) {
    int f = lane + 32 * j;
    float y = (v[j] - mu) * rstd * ln_g[f] + ln_b[f];
    y = fmaxf(y, 0.f);
    hbuf[base + f] = hbuf[base + f] + y;  // add skip (in place)
  }
}

// ---------------------------------------------------------------------------
// Final epilogue: out = acc + hW1*dinv^2 + b1   (one thread per element)
// ---------------------------------------------------------------------------
__global__ void post1_kernel(const float* __restrict__ acc,
                             const float* __restrict__ hw,
                             const float* __restrict__ dinv,
                             const float* __restrict__ b1,
                             float* __restrict__ out, int Nn) {
  int idx = blockIdx.x * blockDim.x + threadIdx.x;
  int total = Nn * 96;
  if (idx >= total) return;
  int n = idx / 96;
  int f = idx - n * 96;
  float di = dinv[n];
  out[idx] = acc[idx] + hw[idx] * di * di + b1[f];
}

// ---------------------------------------------------------------------------
extern "C" void kernel_launch(void* const* d_in, const int* in_sizes, int n_in,
                              void* d_out, int out_size, void* d_ws,
                              size_t ws_size, hipStream_t stream) {
  const float* x = (const float*)d_in[0];
  const long long* ei = (const long long*)d_in[1];  // int64 [2, E]
  const float* ew = (const float*)d_in[2];
  const float* W0 = (const float*)d_in[3];
  const float* b0 = (const float*)d_in[4];
  const float* lng = (const float*)d_in[5];
  const float* lnb = (const float*)d_in[6];
  const float* Wsk = (const float*)d_in[7];
  const float* bsk = (const float*)d_in[8];
  const float* W1 = (const float*)d_in[9];
  const float* b1 = (const float*)d_in[10];

  const int IN = 64, HID = 96;
  int Nn = in_sizes[0] / IN;   // 50000
  int E = in_sizes[2];         // 800000

  const long long* srcI = ei;
  const long long* dstI = ei + E;

  // Workspace layout (floats)
  float* ws = (float*)d_ws;
  float* deg = ws;                               // N
  float* dinv = deg + Nn;                        // N
  float* normv = dinv + Nn;                      // E
  float* buf1 = normv + E;                       // N*96 : xW0, then hW1
  float* buf2 = buf1 + (size_t)Nn * HID;         // N*96 : skip, then h
  float* buf3 = buf2 + (size_t)Nn * HID;         // N*96 : acc0, then acc1

  size_t featN = (size_t)Nn * HID;

  // --- normalization coefficients ----------------------------------------
  fill_kernel<<<2048, 256, 0, stream>>>(deg, 1.0f, (size_t)Nn);   // self loop
  fill_kernel<<<2048, 256, 0, stream>>>(buf3, 0.0f, featN);       // acc0 = 0
  deg_kernel<<<(E + 255) / 256, 256, 0, stream>>>(dstI, ew, deg, E);
  dinv_kernel<<<(Nn + 255) / 256, 256, 0, stream>>>(deg, dinv, Nn);
  norm_kernel<<<(E + 255) / 256, 256, 0, stream>>>(srcI, dstI, ew, dinv, normv, E);

  int mtiles = (Nn + 15) / 16;            // 3125
  int gemm_blocks = (mtiles + 3) / 4;     // 4 waves per block

  // --- layer 0 -------------------------------------------------------------
  gemm_wmma_kernel<64, 96><<<gemm_blocks, 128, 0, stream>>>(x, W0, nullptr, buf1, Nn);
  gemm_wmma_kernel<64, 96><<<gemm_blocks, 128, 0, stream>>>(x, Wsk, bsk, buf2, Nn);
  scatter_kernel<<<(E + 7) / 8, 256, 0, stream>>>(buf1, srcI, dstI, normv, buf3, E);
  post0_kernel<<<(Nn + 7) / 8, 256, 0, stream>>>(buf3, buf1, dinv, b0, lng, lnb, buf2, Nn);

  // --- layer 1 -------------------------------------------------------------
  fill_kernel<<<2048, 256, 0, stream>>>(buf3, 0.0f, featN);       // acc1 = 0
  gemm_wmma_kernel<96, 96><<<gemm_blocks, 128, 0, stream>>>(buf2, W1, nullptr, buf1, Nn);
  scatter_kernel<<<(E + 7) / 8, 256, 0, stream>>>(buf1, srcI, dstI, normv, buf3, E);
  post1_kernel<<<((Nn * 96) + 255) / 256, 256, 0, stream>>>(buf3, buf1, dinv, b1,
                                                            (float*)d_out, Nn);
}